// Sampler_55619826483277
// MI455X (gfx1250) — compile-verified
//
#include <hip/hip_runtime.h>

// ---------------------------------------------------------------------------
// Sampler (temperature + top-k + top-p + gumbel-max) for MI455X / gfx1250.
//
// 1 workgroup (1024 threads = 32 wave32) per batch row.
//  * Selection runs on an order-preserving float->uint key of raw logits
//    (temperature scaling is positive monotone).
//  * top-k threshold: exact 3-pass (11/11/10-bit) histogram radix select.
//  * top-p cutoff:   exact 3-pass histogram descent on fixed-point (2^-42)
//    exp-mass (u64 LDS atomics -> order independent -> deterministic); the
//    total mass S and P = trunc(top_p*S) are derived from pass-0's histogram,
//    eliminating a separate reduction stream.
//  * Histogram streams use CDNA5 async copies: global_load_async_to_lds_b128
//    into wave-private double buffers, s_wait_asynccnt for completion,
//    s_wait_dscnt 0 to close the LDS read-vs-async-write WAR hazard.
//  * Count histograms use wave32 ballot leader aggregation (ds_add_u32).
//  * Argmax phases use vectorized b128 loads + unrolling for memory-level
//    parallelism; 32 waves/block give 8 waves/SIMD of latency hiding.
// ---------------------------------------------------------------------------

#define NTHREADS 1024
#define NWAVES   (NTHREADS / 32)
#define NBINS    2048

__device__ __forceinline__ unsigned orderedU(float f) {
  unsigned b = __float_as_uint(f);
  return (b & 0x80000000u) ? ~b : (b | 0x80000000u);  // monotone with float order
}

// exp((l - lmax) * invT) in 2^-42 fixed point; value <= 2^42, V*2^42 < 2^59.
__device__ __forceinline__ unsigned long long fixedExp(float l, float lmax, float invT) {
  float e = expf((l - lmax) * invT);
  return (unsigned long long)(e * 4398046511104.0f);  // * 2^42
}

// Wave32 leader-aggregated LDS count histogram add (collapses same-bin atomics).
__device__ __forceinline__ void waveHistCount(unsigned* histc, unsigned bin, bool active) {
  unsigned long long am = __ballot(active ? 1 : 0);
  int lane = (int)(threadIdx.x & 31);
  while (am) {
    int leader = __ffsll(am) - 1;
    unsigned lbin = (unsigned)__shfl((int)bin, leader);
    unsigned long long match = __ballot((active && bin == lbin) ? 1 : 0);
    if (lane == leader) atomicAdd(&histc[lbin], (unsigned)__popcll(match));
    am &= ~match;
  }
}

// Stream one row: full tiles via async global->LDS (double-buffered, wave
// private so no block barriers in the loop), tail via direct loads.
template <typename F>
__device__ __forceinline__ void streamRow(const float* __restrict__ rowPtr, int V,
                                          int wave, int lane, int tid,
                                          float* stageAll, F f) {
  float* base = stageAll + wave * 256;          // 2 buffers x 128 floats per wave
  const int TILE = NWAVES * 128;                // elements per block-tile
  const int nFull = V / TILE;

  auto issue = [&](int t, int buf) {
    unsigned ldsAddr = (unsigned)(unsigned long long)(base + buf * 128 + lane * 4);
    int byteOff = (t * TILE + wave * 128 + lane * 4) * 4;
    // s_wait_dscnt 0: prior ds reads of this buffer must retire before the
    // async engine overwrites it (WAR on LDS).
    asm volatile("s_wait_dscnt 0x0\n\t"
                 "global_load_async_to_lds_b128 %0, %1, %2"
                 :: "v"(ldsAddr), "v"(byteOff), "s"(rowPtr) : "memory");
  };
  auto proc = [&](int buf) {
    const float* p = base + buf * 128 + lane * 4;
    f(p[0]); f(p[1]); f(p[2]); f(p[3]);
  };

  if (nFull >= 2) {
    issue(0, 0); issue(1, 1);
    for (int t = 0; t < nFull; ++t) {
      if (t < nFull - 1) asm volatile("s_wait_asynccnt 0x1" ::: "memory");
      else               asm volatile("s_wait_asynccnt 0x0" ::: "memory");
      proc(t & 1);
      if (t + 2 < nFull) issue(t + 2, t & 1);
    }
  } else if (nFull == 1) {
    issue(0, 0);
    asm volatile("s_wait_asynccnt 0x0" ::: "memory");
    proc(0);
  }
  for (int i = nFull * TILE + tid; i < V; i += NTHREADS) f(rowPtr[i]);
}

// Fixed-tree (deterministic) block argmax over sVal/sIdx, first-index ties.
__device__ __forceinline__ void blockArgmaxReduce(float* sV, int* sI, int tid) {
  __syncthreads();
  for (int s = NTHREADS >> 1; s > 0; s >>= 1) {
    if (tid < s) {
      float ov = sV[tid + s]; int oi = sI[tid + s];
      if (ov > sV[tid] || (ov == sV[tid] && oi < sI[tid])) { sV[tid] = ov; sI[tid] = oi; }
    }
    __syncthreads();
  }
}

__global__ __launch_bounds__(NTHREADS) void sampler_kernel(
    const float* __restrict__ logits, const float* __restrict__ temperature,
    const float* __restrict__ top_p, const int* __restrict__ top_k,
    const float* __restrict__ gumbel_u, int* __restrict__ out, int V) {
  __shared__ unsigned long long histU[NBINS];             // 16 KB (u32 alias for counts)
  __shared__ __align__(16) float stage[NWAVES * 2 * 128]; // 32 KB async staging
  __shared__ float sVal[NTHREADS];                        // 4 KB
  __shared__ int   sIdx[NTHREADS];                        // 4 KB
  __shared__ unsigned sPrefix, sNeedK;
  __shared__ unsigned long long sA, sP;
  __shared__ int sDoneAll;
  unsigned* histc = reinterpret_cast<unsigned*>(histU);

  const int row  = blockIdx.x;
  const int tid  = threadIdx.x;
  const int wave = tid >> 5, lane = tid & 31;
  const float* rowL = logits   + (size_t)row * V;
  const float* rowG = gumbel_u + (size_t)row * V;

  const float temp  = temperature[row];
  const float safeT = (temp < 1e-5f) ? 1.0f : temp;
  const float invT  = 1.0f / safeT;
  const int   k     = top_k[row];

  // Radix pass tables: key bits 31..21, 20..10, 9..0.
  const int      shifts[3] = {21, 10, 0};
  const unsigned bmaskA[3] = {0x7FFu, 0x7FFu, 0x3FFu};
  const unsigned pmaskA[3] = {0u, 0xFFE00000u, 0xFFFFFC00u};
  const int      nbA[3]    = {2048, 2048, 1024};

  const int n4 = ((V & 3) == 0) ? (V >> 2) : 0;   // b128 path requires V%4==0
  const float4* rowL4 = (const float4*)rowL;

  // ---- Phase 1: max & argmax of logits (greedy token) -------------------
  float bv = -INFINITY; int bi = 0x7FFFFFFF;
#pragma unroll 4
  for (int i4 = tid; i4 < n4; i4 += NTHREADS) {
    float4 v = rowL4[i4];
    const int b0 = i4 << 2;
    if (v.x > bv || (v.x == bv && b0 + 0 < bi)) { bv = v.x; bi = b0 + 0; }
    if (v.y > bv || (v.y == bv && b0 + 1 < bi)) { bv = v.y; bi = b0 + 1; }
    if (v.z > bv || (v.z == bv && b0 + 2 < bi)) { bv = v.z; bi = b0 + 2; }
    if (v.w > bv || (v.w == bv && b0 + 3 < bi)) { bv = v.w; bi = b0 + 3; }
  }
  for (int i = (n4 << 2) + tid; i < V; i += NTHREADS) {
    float v = rowL[i];
    if (v > bv || (v == bv && i < bi)) { bv = v; bi = i; }
  }
  sVal[tid] = bv; sIdx[tid] = bi;
  blockArgmaxReduce(sVal, sIdx, tid);
  const float lmax   = sVal[0];
  const int   greedy = sIdx[0];
  __syncthreads();

  // ---- Phase 2: exact kth-largest key (top-k threshold) -----------------
  unsigned kthOu = 0u;
  if (k > 0) {
    if (tid == 0) { sPrefix = 0u; sNeedK = (unsigned)((k < V) ? k : V); }
    __syncthreads();
    for (int p = 0; p < 3; ++p) {
      for (int i = tid; i < NBINS; i += NTHREADS) histc[i] = 0u;
      __syncthreads();
      const unsigned pref = sPrefix, pm = pmaskA[p], bm = bmaskA[p];
      const int sh = shifts[p];
      streamRow(rowL, V, wave, lane, tid, stage, [&](float l) {
        unsigned u = orderedU(l);
        waveHistCount(histc, (u >> sh) & bm, (u & pm) == pref);
      });
      __syncthreads();
      if (tid == 0) {
        unsigned need = sNeedK, acc = 0u, sel = 0u;
        for (int b = nbA[p] - 1; b >= 0; --b) {
          unsigned c = histc[b];
          if (acc + c >= need) { sel = (unsigned)b; break; }
          acc += c;
        }
        sNeedK  = need - acc;
        sPrefix = pref | (sel << sh);
      }
      __syncthreads();
    }
    kthOu = sPrefix;           // exact ordered key of kth largest
    __syncthreads();
  }

  // ---- Phase 3: top-p cutoff via exp-mass radix descent -----------------
  // Pass 0's histogram doubles as the total-mass reduction: S = sum(bins),
  // P = trunc(top_p * S); both exact in u64 fixed point => deterministic.
  if (tid == 0) { sPrefix = 0u; sA = 0ull; sDoneAll = 0; }
  __syncthreads();
  for (int p = 0; p < 3; ++p) {
    if (sDoneAll) break;                       // uniform (read after barrier)
    for (int i = tid; i < NBINS; i += NTHREADS) histU[i] = 0ull;
    __syncthreads();
    const unsigned pref = sPrefix, pm = pmaskA[p], bm = bmaskA[p];
    const int sh = shifts[p];
    streamRow(rowL, V, wave, lane, tid, stage, [&](float l) {
      unsigned u = orderedU(l);
      if (u >= kthOu && (u & pm) == pref)
        atomicAdd(&histU[(u >> sh) & bm], fixedExp(l, lmax, invT));
    });
    __syncthreads();
    if (tid == 0) {
      if (p == 0) {            // fused total-mass reduction
        unsigned long long S = 0ull;
        for (int b = 0; b < nbA[0]; ++b) S += histU[b];
        sP = (unsigned long long)((double)top_p[row] * (double)S);
      }
      unsigned long long A = sA, P = sP; unsigned sel = 0u; int found = 0;
      for (int b = nbA[p] - 1; b >= 0; --b) {
        unsigned long long h = histU[b];
        if (A + h > P) { sel = (unsigned)b; found = 1; break; }  // crossing bin
        A += h;                                // whole bin kept
      }
      sA = A;
      if (found) sPrefix = pref | (sel << sh);
      else { sDoneAll = 1; sPrefix = kthOu; }  // total <= P: keep all kept
    }
    __syncthreads();
  }
  const unsigned cutoff = sPrefix;             // keep key >= cutoff
  __syncthreads();

  // ---- Phase 4: Gumbel-max over surviving tokens ------------------------
  float by = -INFINITY; int bj = 0x7FFFFFFF;
#pragma unroll 2
  for (int i4 = tid; i4 < n4; i4 += NTHREADS) {
    float4 v = rowL4[i4];
    const int b0 = i4 << 2;
    float lv[4] = {v.x, v.y, v.z, v.w};
#pragma unroll
    for (int j = 0; j < 4; ++j) {
      float l = lv[j];
      if (orderedU(l) >= cutoff) {             // rare: <= ~top-k survivors
        float gu = rowG[b0 + j];
        float g  = -logf(-logf(gu + 1e-20f) + 1e-20f);
        float y  = (l - lmax) * invT + g;      // constant shift: argmax-invariant
        if (y > by || (y == by && b0 + j < bj)) { by = y; bj = b0 + j; }
      }
    }
  }
  for (int i = (n4 << 2) + tid; i < V; i += NTHREADS) {
    float l = rowL[i];
    if (orderedU(l) >= cutoff) {
      float gu = rowG[i];
      float g  = -logf(-logf(gu + 1e-20f) + 1e-20f);
      float y  = (l - lmax) * invT + g;
      if (y > by || (y == by && i < bj)) { by = y; bj = i; }
    }
  }
  sVal[tid] = by; sIdx[tid] = bj;
  blockArgmaxReduce(sVal, sIdx, tid);
  if (tid == 0) out[row] = (temp < 1e-5f) ? greedy : sIdx[0];
}

extern "C" void kernel_launch(void* const* d_in, const int* in_sizes, int n_in,
                              void* d_out, int out_size, void* d_ws, size_t ws_size,
                              hipStream_t stream) {
  const float* logits      = (const float*)d_in[0];
  const float* temperature = (const float*)d_in[1];
  const float* top_p       = (const float*)d_in[2];
  const int*   top_k       = (const int*)d_in[3];
  const float* gumbel_u    = (const float*)d_in[4];
  const int B = in_sizes[1];
  const int V = in_sizes[0] / B;
  sampler_kernel<<<B, NTHREADS, 0, stream>>>(logits, temperature, top_p, top_k,
                                             gumbel_u, (int*)d_out, V);
}